// PredictTargets_1580547972278
// MI455X (gfx1250) — compile-verified
//
#include <hip/hip_runtime.h>

// Problem constants from the reference: nt=ne=48, b=32, vs=512.
#define NT 48
#define NE 48
#define NB 32
#define VS 512
#define ROWS (NT * NB)   // 1536 flattened (t,b) rows; same count for (n,b)

typedef __attribute__((ext_vector_type(2))) float v2f;
typedef __attribute__((ext_vector_type(8))) float v8f;

#if __has_builtin(__builtin_amdgcn_tanhf)
#define TANHF(x) __builtin_amdgcn_tanhf(x)   // gfx1250 v_tanh_f32
#else
#define TANHF(x) tanhf(x)
#endif

// ---------------------------------------------------------------------------
// Stage 1: TA[row,h] = sum_v X[row,v] * W1[h, v(+coloff)]  (+ b1[h] for TA)
// One wave (32 threads) computes one 16x16 f32 output tile via
// V_WMMA_F32_16X16X4_F32, streaming K in steps of 4.
//
// f32 16x4 A fragment layout (ISA 7.12.2): lane l, vgpr j holds
//   A[M = l%16, K = (l/16)*2 + j];  B fragment mirrors with N = l%16.
// Both A (row-major, K contiguous) and B (= W1 row h, K contiguous) are
// loaded as a single b64 per lane per K-step.
// ---------------------------------------------------------------------------
__global__ __launch_bounds__(32) void mlp_gemm_wmma(
    const float* __restrict__ targets, const float* __restrict__ embeddings,
    const float* __restrict__ W1, const float* __restrict__ b1,
    float* __restrict__ TA, float* __restrict__ EB)
{
    const int h0    = blockIdx.x * 16;   // output column tile (hidden dim)
    const int row0  = blockIdx.y * 16;   // output row tile (flattened t*32+b)
    const int which = blockIdx.z;        // 0: targets/W1a (+b1)  1: embeddings/W1b
    const int lane  = threadIdx.x;
    const int m     = lane & 15;
    const int half  = lane >> 4;

    const float* __restrict__ A =
        (which ? embeddings : targets) + (size_t)(row0 + m) * VS;
    // W1 is (512, 1024) row-major; block a = cols [0,512), block b = [512,1024)
    const float* __restrict__ B =
        W1 + (size_t)(h0 + m) * (2 * VS) + (which ? VS : 0);

    v8f acc = {};
#pragma unroll 8
    for (int k = 0; k < VS; k += 4) {
        v2f a = *(const v2f*)(A + k + half * 2);
        v2f b = *(const v2f*)(B + k + half * 2);
        // 8 args: (neg_a, A, neg_b, B, c_mod, C, reuse_a, reuse_b)
        acc = __builtin_amdgcn_wmma_f32_16x16x4_f32(
            false, a, false, b, (short)0, acc, false, false);
    }

    // D layout: vgpr r, lanes 0-15 -> M=r, N=lane; lanes 16-31 -> M=r+8.
    const int   h    = h0 + m;
    const float bias = (which == 0) ? b1[h] : 0.0f;  // fold b1 into TA only
    float* __restrict__ C = which ? EB : TA;
#pragma unroll
    for (int r = 0; r < 8; ++r) {
        const int row = row0 + r + half * 8;
        C[(size_t)row * VS + h] = acc[r] + bias;
    }
}

// ---------------------------------------------------------------------------
// Stage 2: out[t,n,b] = sum_h tanh(TA[t,b,h] + EB[n,b,h]) * W2[h] + b2
// One block per (t,n) pair; 8 waves, each wave covers 4 batch rows.
// Lanes stride h with float4 (b128) loads; W2 is preloaded into registers
// once per wave; tanh lowers to v_tanh_f32; wave32 shuffle reduction.
// ---------------------------------------------------------------------------
__global__ __launch_bounds__(256) void pairwise_tanh_dot(
    const float* __restrict__ TA, const float* __restrict__ EB,
    const float* __restrict__ W2, const float* __restrict__ b2,
    float* __restrict__ out)
{
    const int tn   = blockIdx.x;          // 0 .. 48*48-1
    const int t    = tn / NE;
    const int n    = tn % NE;
    const int wave = threadIdx.x >> 5;
    const int lane = threadIdx.x & 31;

    // Preload W2 (512 floats) into registers: 4 float4 per lane.
    float4 w2r[4];
    const float4* __restrict__ W24 = (const float4*)W2;
#pragma unroll
    for (int i = 0; i < 4; ++i) w2r[i] = W24[i * 32 + lane];
    const float bias2 = b2[0];

    for (int b = wave; b < NB; b += 8) {
        const float4* __restrict__ pta =
            (const float4*)(TA + ((size_t)t * NB + b) * VS);
        const float4* __restrict__ peb =
            (const float4*)(EB + ((size_t)n * NB + b) * VS);
        float acc = 0.0f;
#pragma unroll
        for (int i = 0; i < 4; ++i) {
            const float4 a = pta[i * 32 + lane];
            const float4 e = peb[i * 32 + lane];
            const float4 w = w2r[i];
            acc = fmaf(TANHF(a.x + e.x), w.x, acc);
            acc = fmaf(TANHF(a.y + e.y), w.y, acc);
            acc = fmaf(TANHF(a.z + e.z), w.z, acc);
            acc = fmaf(TANHF(a.w + e.w), w.w, acc);
        }
        // wave32 reduction
#pragma unroll
        for (int off = 16; off > 0; off >>= 1)
            acc += __shfl_down(acc, off, 32);
        if (lane == 0) out[(size_t)tn * NB + b] = acc + bias2;
    }
}

extern "C" void kernel_launch(void* const* d_in, const int* in_sizes, int n_in,
                              void* d_out, int out_size, void* d_ws, size_t ws_size,
                              hipStream_t stream) {
    const float* targets    = (const float*)d_in[0];  // (48,32,512)
    const float* embeddings = (const float*)d_in[1];  // (48,32,512)
    const float* W1         = (const float*)d_in[2];  // (512,1024)
    const float* b1         = (const float*)d_in[3];  // (512,)
    const float* W2         = (const float*)d_in[4];  // (1,512)
    const float* b2         = (const float*)d_in[5];  // (1,)
    float*       out        = (float*)d_out;          // (48,48,32)

    float* TA = (float*)d_ws;             // 1536*512 floats = 3 MB
    float* EB = TA + (size_t)ROWS * VS;   // next 3 MB

    dim3 g1(VS / 16, ROWS / 16, 2);       // (32, 96, 2) tiles, 1 wave each
    mlp_gemm_wmma<<<g1, 32, 0, stream>>>(targets, embeddings, W1, b1, TA, EB);

    pairwise_tanh_dot<<<NT * NE, 256, 0, stream>>>(TA, EB, W2, b2, out);
}